// ChamferLoss_76802605187598
// MI455X (gfx1250) — compile-verified
//
#include <hip/hip_runtime.h>

// CDNA5 / gfx1250 Chamfer distance via V_WMMA_F32_16X16X4_F32.
// Augmented K=4 trick: A' = [-2ax,-2ay,-2az, 1], B' = [bx,by,bz, ||b||^2]
//   => C_ij = -2 a.b + bb_j straight from the matrix unit (no VALU adds).
// sum_i min_j d = sum_i aa_i + sum_i min_j C_ij   (aa decoupled).
//  - B' pre-transformed into native WMMA B-layout in LDS once per point.
//  - 2 A-tiles (32 rows) per wave: 2 WMMAs amortize each B-tile ds_load_b64.
//  - Both chamfer directions fused into one launch via blockIdx.z.

typedef __attribute__((ext_vector_type(2))) float v2f;
typedef __attribute__((ext_vector_type(8))) float v8f;

#define WAVES      8
#define BLOCK      (WAVES * 32)
#define TILE_ROWS  16
#define ROWS_PER_WAVE (2 * TILE_ROWS)
#define CHUNK      2048              // target points resident in LDS per pass
#define NTILES     (CHUNK / 16)      // 128 B-tiles per chunk
#define BATCH      4
#define NPTS       8192

__global__ void chamfer_zero_kernel(float* out) { out[0] = 0.0f; }

__global__ void __launch_bounds__(BLOCK)
chamfer_kernel(const float* __restrict__ pred,
               const float* __restrict__ gt,
               float* __restrict__ out,
               float scale)
{
    // B' in native WMMA B-matrix layout:
    //   sBW[t][lane]: lanes 0-15  = {bx, by}      (K0,K1) of col N=lane
    //                 lanes 16-31 = {bz, ||b||^2} (K2,K3) of col N=lane-16
    __shared__ float2 sBW[NTILES][32];   // 32 KB

    // Direction select: z=0 -> pred->gt, z=1 -> gt->pred.
    const float* __restrict__ A  = (blockIdx.z == 0) ? pred : gt;
    const float* __restrict__ Bp = (blockIdx.z == 0) ? gt   : pred;

    const int batch = blockIdx.y;
    A  += (size_t)batch * NPTS * 3;
    Bp += (size_t)batch * NPTS * 3;

    const int tid  = threadIdx.x;
    const int lane = tid & 31;
    const int wave = tid >> 5;
    const int col  = lane & 15;
    const bool hi  = lane >= 16;

    // ---- Two 16-row A' tiles per wave, in WMMA A-matrix layout ----
    // lanes 0-15: {-2x, -2y} (K0,K1); lanes 16-31: {-2z, 1} (K2,K3).
    const int i0 = (blockIdx.x * WAVES + wave) * ROWS_PER_WAVE;
    const int r0 = i0 + col;
    const int r1 = i0 + TILE_ROWS + col;

    const float ax0 = A[3 * r0 + 0], ay0 = A[3 * r0 + 1], az0 = A[3 * r0 + 2];
    const float ax1 = A[3 * r1 + 0], ay1 = A[3 * r1 + 1], az1 = A[3 * r1 + 2];
    v2f areg0, areg1;
    areg0.x = hi ? (-2.0f * az0) : (-2.0f * ax0);
    areg0.y = hi ? 1.0f          : (-2.0f * ay0);
    areg1.x = hi ? (-2.0f * az1) : (-2.0f * ax1);
    areg1.y = hi ? 1.0f          : (-2.0f * ay1);
    const float aa0 = hi ? 0.0f : (ax0 * ax0 + ay0 * ay0 + az0 * az0);
    const float aa1 = hi ? 0.0f : (ax1 * ax1 + ay1 * ay1 + az1 * az1);

    float rmin0[8], rmin1[8];
#pragma unroll
    for (int v = 0; v < 8; ++v) { rmin0[v] = 3.4e38f; rmin1[v] = 3.4e38f; }

    for (int c0 = 0; c0 < NPTS; c0 += CHUNK) {
        __syncthreads();
        // Chunk fill: layout transform + bb done ONCE per point (not per wave).
        for (int p = tid; p < CHUNK; p += BLOCK) {
            const int j = c0 + p;
            const float bx = Bp[3 * j + 0];
            const float by = Bp[3 * j + 1];
            const float bz = Bp[3 * j + 2];
            const int t = p >> 4, c = p & 15;
            sBW[t][c]      = make_float2(bx, by);                          // K0,K1
            sBW[t][c + 16] = make_float2(bz, bx * bx + by * by + bz * bz); // K2,K3
        }
        __syncthreads();

        for (int t = 0; t < NTILES; ++t) {
            const float2 bw = sBW[t][lane];       // ds_load_b64, conflict-free
            v2f breg; breg.x = bw.x; breg.y = bw.y;

            v8f c0r = {}, c1r = {};
            c0r = __builtin_amdgcn_wmma_f32_16x16x4_f32(
                      false, areg0, false, breg, (short)0, c0r, false, false);
            c1r = __builtin_amdgcn_wmma_f32_16x16x4_f32(
                      false, areg1, false, breg, (short)0, c1r, false, false);

            // C already equals -2 a.b + bb : pure min-accumulate.
#pragma unroll
            for (int v = 0; v < 8; ++v) {
                rmin0[v] = fminf(rmin0[v], c0r[v]);
                rmin1[v] = fminf(rmin1[v], c1r[v]);
            }
        }
    }

    // ---- Row-min reduction across the 16 columns in each half-wave ----
#pragma unroll
    for (int v = 0; v < 8; ++v) {
#pragma unroll
        for (int m = 1; m < 16; m <<= 1) {
            rmin0[v] = fminf(rmin0[v], __shfl_xor(rmin0[v], m, 32));
            rmin1[v] = fminf(rmin1[v], __shfl_xor(rmin1[v], m, 32));
        }
    }
    float s = 0.0f;
#pragma unroll
    for (int v = 0; v < 8; ++v) s += rmin0[v] + rmin1[v];

    // Valid row-min sums live on lanes 0 (rows M=0..7) and 16 (rows M=8..15)
    // of each tile; aa valid on lanes 0..15. Wave-sum everything.
    float contrib = (hi ? 0.0f : (aa0 + aa1)) +
                    (((lane == 0) || (lane == 16)) ? s : 0.0f);
#pragma unroll
    for (int m = 1; m < 32; m <<= 1) contrib += __shfl_xor(contrib, m, 32);

    if (lane == 0)
        unsafeAtomicAdd(out, contrib * scale);
}

extern "C" void kernel_launch(void* const* d_in, const int* in_sizes, int n_in,
                              void* d_out, int out_size, void* d_ws, size_t ws_size,
                              hipStream_t stream) {
    (void)in_sizes; (void)n_in; (void)out_size; (void)d_ws; (void)ws_size;
    const float* pred = (const float*)d_in[0];
    const float* gt   = (const float*)d_in[1];
    float* out = (float*)d_out;

    chamfer_zero_kernel<<<1, 1, 0, stream>>>(out);

    // grid: 32 row-groups x 4 batches x 2 directions = 256 blocks
    const dim3 grid(NPTS / (WAVES * ROWS_PER_WAVE), BATCH, 2);
    const float scale = 1.0f / (float)(BATCH * NPTS);
    chamfer_kernel<<<grid, BLOCK, 0, stream>>>(pred, gt, out, scale);
}